// Discriminator_77034533421678
// MI455X (gfx1250) — compile-verified
//
#include <hip/hip_runtime.h>
#include <hip/hip_bf16.h>
#include <stdint.h>

// ---------------------------------------------------------------------------
// AdderNet discriminator forward for MI455X (gfx1250, wave32).
//   adder2d:  out = -sum |patch - w|   (L1 distance; NOT a matmul -> VALU)
//   BN (batch stats) + LeakyReLU(0.2), final sigmoid.
// CDNA5 paths used:
//   * TENSOR_LOAD_TO_LDS (TDM) with row-padding + iterate mode to DMA the
//     whole padded input image into LDS (one descriptor per block).
//   * global_load_async_to_lds_b32 (ASYNCcnt) for the weight tile.
//   * s_wait_tensorcnt / s_wait_asynccnt split counters.
// Inner loop: v_sub_f32 + v_add_f32 with |src| modifier, TW outputs/lane.
// ---------------------------------------------------------------------------

#define EPS 1e-5f
#define NEG_SLOPE 0.2f

__device__ __forceinline__ uint32_t lds_off_u32(const void* p) {
  // Flat shared-aperture pointer: addr[31:0] is the LDS byte offset.
  return (uint32_t)(uintptr_t)p;
}

// --- per-lane async global->LDS copy (4 bytes), tracked by ASYNCcnt --------
__device__ __forceinline__ void async_copy_b32(void* lds_dst, const float* gsrc) {
  uint32_t dst = lds_off_u32(lds_dst);
  asm volatile("global_load_async_to_lds_b32 %0, %1, off"
               :: "v"(dst), "v"(gsrc)
               : "memory");
}

constexpr int ilog2(int v) { int r = 0; while (v > 1) { v >>= 1; ++r; } return r; }

// ---------------------------------------------------------------------------
// adder2d kernel: stride=2, pad=1 (all 5 layers).
// Grid: (N, CO/CO_T). Block: 256 threads (8 wave32s).
// LDS: padded input image  CI*(HI+2)*(WI+2) floats  +  CO_T*CI*K*K weights.
// ---------------------------------------------------------------------------
template <int CI, int K, int HI, int WI, int HO, int WO, int CO, int CO_T, int TW>
__global__ void adder2d_kernel(const float* __restrict__ in,
                               const float* __restrict__ W,
                               float* __restrict__ out) {
  constexpr int PH = HI + 2;
  constexpr int PW = WI + 2;
  constexpr int NW = WO / TW;             // horizontal chunks per row
  constexpr int ITEMS = CO_T * HO * NW;   // work items per block
  constexpr int PAD_IV = ilog2(WI) - 1;   // pad-interval enum: 2<<k dwords = WI

  extern __shared__ float smem[];
  float* in_s = smem;                     // CI * PH * PW
  float* w_s  = smem + CI * PH * PW;      // CO_T * CI * K * K

  const int tid = threadIdx.x;
  const int nt  = blockDim.x;
  const int n   = blockIdx.x;
  const int co0 = blockIdx.y * CO_T;
  const float* gin = in + (size_t)n * CI * HI * WI;

  // ---- TDM: DMA the interior of the input image into padded LDS ----------
  // One 2D tile (HI rows x WI dwords) per channel; pad_amount=2 dwords after
  // every WI dwords gives LDS row stride PW; iterate mode steps channels by
  // lds_addr_increment = PH*PW dwords (skipping the 2 halo rows per channel).
  if (tid == 0) {
    typedef uint32_t u32x4 __attribute__((ext_vector_type(4)));
    typedef uint32_t u32x8 __attribute__((ext_vector_type(8)));
    const uint64_t ga = (uint64_t)(uintptr_t)gin;
    const uint32_t la = lds_off_u32(in_s) + (uint32_t)(PW + 1) * 4u;

    u32x4 g0; u32x8 g1; u32x4 g2; u32x4 g3;
    // group0: count=1 | lds_addr | global_addr[56:0] | type=2 ("image")
    g0[0] = 1u;
    g0[1] = la;
    g0[2] = (uint32_t)ga;
    g0[3] = (uint32_t)((ga >> 32) & 0x01FFFFFFu) | (2u << 30);
    // group1: mask=0, data_size=4B(2), iterate_en(19), pad_en(20),
    //         pad_interval(24:22), pad_amount=2 dwords (enum 1, 31:25)
    g1[0] = (2u << 16) | (1u << 19) | (1u << 20) |
            ((uint32_t)PAD_IV << 22) | (1u << 25);
    g1[1] = ((uint32_t)WI & 0xFFFFu) << 16;                    // tensor_dim0 lo
    g1[2] = ((uint32_t)WI >> 16) | (((uint32_t)HI) << 16);     // dim0 hi|dim1 lo
    g1[3] = ((uint32_t)WI << 16);                              // dim1 hi|tile_dim0
    g1[4] = (uint32_t)HI;                                      // tile_dim1 (|tile_dim2=0)
    g1[5] = (uint32_t)WI;                                      // tensor_dim0_stride lo
    g1[6] = 0u;                                                // stride0 hi | stride1 lo
    g1[7] = 0u;                                                // stride1 hi
    // group2 (iterate mode): lds/global increments per channel + count
    g2[0] = 0u;                                                // tensor_dim2
    g2[1] = (uint32_t)(PH * PW);            // lds_addr_increment (dwords)
    g2[2] = (uint32_t)(HI * WI);            // global_addr_increment (dwords)
    g2[3] = ((uint32_t)(CI - 1)) << 16;     // iterate_count (CI iterations)
    g3[0] = 0u; g3[1] = 0u; g3[2] = 0u; g3[3] = 0u;

    asm volatile("tensor_load_to_lds %0, %1, %2, %3"
                 :: "s"(g0), "s"(g1), "s"(g2), "s"(g3)
                 : "memory");
  }

  // ---- zero the padding halo (top/bottom rows, left/right cols) -----------
  // Disjoint from the TDM-written interior (TDM *skips* the pad dwords).
  constexpr int HALO = 2 * PW + 2 * HI;    // per input channel
  for (int e = tid; e < CI * HALO; e += nt) {
    const int ci = e / HALO;
    const int r  = e % HALO;
    float* base = in_s + ci * PH * PW;
    if (r < PW)            base[r] = 0.f;                        // top row
    else if (r < 2 * PW)   base[(PH - 1) * PW + (r - PW)] = 0.f; // bottom row
    else {
      const int rr = r - 2 * PW;
      const int row = rr >> 1;
      base[(row + 1) * PW + ((rr & 1) ? (PW - 1) : 0)] = 0.f;    // side cols
    }
  }

  // ---- async-stage the weight tile (ASYNCcnt path) ------------------------
  const float* gw = W + (size_t)co0 * CI * K * K;
  for (int e = tid; e < CO_T * CI * K * K; e += nt) {
    async_copy_b32(&w_s[e], gw + e);
  }

  __builtin_amdgcn_s_wait_tensorcnt(0);   // wave 0 drains the TDM op
  __builtin_amdgcn_s_wait_asynccnt(0);    // each wave drains its async fills
  __syncthreads();                        // halo stores + fills visible to all

  // ---- compute: each item = (co, ho, w-chunk of TW outputs) ---------------
  for (int item = tid; item < ITEMS; item += nt) {
    const int co  = item / (HO * NW);
    const int rem = item % (HO * NW);
    const int ho  = rem / NW;
    const int w0  = (rem % NW) * TW;

    float acc[TW];
#pragma unroll
    for (int t = 0; t < TW; ++t) acc[t] = 0.f;

    const float* wp = &w_s[co * CI * K * K];
    for (int ci = 0; ci < CI; ++ci) {
      const float* ibase = &in_s[ci * PH * PW + (ho * 2) * PW + (w0 * 2)];
#pragma unroll
      for (int kh = 0; kh < K; ++kh) {
        const float* row = ibase + kh * PW;
        const float* wr  = wp + (ci * K + kh) * K;
#pragma unroll
        for (int kw = 0; kw < K; ++kw) {
          const float wv = wr[kw];
#pragma unroll
          for (int t = 0; t < TW; ++t) {
            // v_sub_f32 + v_add_f32 with |src| modifier (2 VALU / element)
            acc[t] += __builtin_fabsf(row[2 * t + kw] - wv);
          }
        }
      }
    }

    float* op = out + (((size_t)n * CO + (co0 + co)) * HO + ho) * WO + w0;
#pragma unroll
    for (int t = 0; t < TW; ++t) op[t] = -acc[t];
  }
}

// ---------------------------------------------------------------------------
// BN batch statistics: one block per channel, deterministic tree reduction.
// Writes ss[c] = g*rsqrt(var+eps), ss[C+c] = b - mean*scale.
// ---------------------------------------------------------------------------
__global__ void bn_stats_kernel(const float* __restrict__ x,
                                const float* __restrict__ g,
                                const float* __restrict__ b,
                                float* __restrict__ ss,
                                int N, int C, int HW) {
  __shared__ float r1[256];
  __shared__ float r2[256];
  const int c = blockIdx.x, tid = threadIdx.x, nt = blockDim.x;

  float s1 = 0.f, s2 = 0.f;
  for (int n = 0; n < N; ++n) {
    const float* p = x + ((size_t)n * C + c) * HW;
    for (int i = tid; i < HW; i += nt) {
      const float v = p[i];
      s1 += v;
      s2 += v * v;
    }
  }
  r1[tid] = s1;
  r2[tid] = s2;
  __syncthreads();
  for (int off = nt >> 1; off > 0; off >>= 1) {
    if (tid < off) { r1[tid] += r1[tid + off]; r2[tid] += r2[tid + off]; }
    __syncthreads();
  }
  if (tid == 0) {
    const float inv = 1.0f / ((float)N * (float)HW);
    const float m   = r1[0] * inv;
    const float var = r2[0] * inv - m * m;
    const float sc  = g[c] * rsqrtf(var + EPS);
    ss[c]     = sc;
    ss[C + c] = b[c] - m * sc;
  }
}

// BN apply + LeakyReLU, in place.
__global__ void bn_apply_lrelu_kernel(float* __restrict__ x,
                                      const float* __restrict__ ss,
                                      int C, int HW, long total) {
  const long idx = (long)blockIdx.x * blockDim.x + threadIdx.x;
  if (idx >= total) return;
  const int c = (int)((idx / HW) % C);
  float v = x[idx] * ss[c] + ss[C + c];
  x[idx] = (v >= 0.f) ? v : NEG_SLOPE * v;
}

// BN apply + sigmoid -> d_out.
__global__ void bn_apply_sigmoid_kernel(const float* __restrict__ x,
                                        const float* __restrict__ ss,
                                        float* __restrict__ out,
                                        int C, int HW, long total) {
  const long idx = (long)blockIdx.x * blockDim.x + threadIdx.x;
  if (idx >= total) return;
  const int c = (int)((idx / HW) % C);
  const float v = x[idx] * ss[c] + ss[C + c];
  out[idx] = 1.0f / (1.0f + expf(-v));
}

// ---------------------------------------------------------------------------
// Host orchestration.
// ---------------------------------------------------------------------------
extern "C" void kernel_launch(void* const* d_in, const int* in_sizes, int n_in,
                              void* d_out, int out_size, void* d_ws, size_t ws_size,
                              hipStream_t stream) {
  const float* x  = (const float*)d_in[0];
  const float* W1 = (const float*)d_in[1];
  const float* g1 = (const float*)d_in[2];
  const float* b1 = (const float*)d_in[3];
  const float* W2 = (const float*)d_in[4];
  const float* g2 = (const float*)d_in[5];
  const float* b2 = (const float*)d_in[6];
  const float* W3 = (const float*)d_in[7];
  const float* g3 = (const float*)d_in[8];
  const float* b3 = (const float*)d_in[9];
  const float* W4 = (const float*)d_in[10];
  const float* g4 = (const float*)d_in[11];
  const float* b4 = (const float*)d_in[12];
  const float* W5 = (const float*)d_in[13];
  const float* g5 = (const float*)d_in[14];
  const float* b5 = (const float*)d_in[15];
  float* out = (float*)d_out;

  const int N = in_sizes[0] / (128 * 128); // 256

  // Workspace: ping-pong buffers + BN scale/shift scratch (~100.7 MB).
  char* ws = (char*)d_ws;
  float* bufA  = (float*)ws;                                   // >= N*16*64*64 f
  float* bufB  = (float*)(ws + (size_t)N * 16 * 64 * 64 * 4);  // >= N*32*32*32 f
  float* stats = (float*)(ws + (size_t)N * 16 * 64 * 64 * 4
                             + (size_t)N * 32 * 32 * 32 * 4);  // 2*128 floats

  const dim3 blk(256);

  // LDS budgets (bytes): padded input image + weight tile, all <= 320 KB.
  constexpr size_t smem1 = (1   * 130 * 130 + 16  * 1   * 16) * 4; //  68,624
  constexpr size_t smem2 = (16  * 66  * 66  + 32  * 16  * 16) * 4; // 311,552
  constexpr size_t smem3 = (32  * 34  * 34  + 64  * 32  * 9 ) * 4; // 221,696
  constexpr size_t smem4 = (64  * 18  * 18  + 32  * 64  * 16) * 4; // 214,016
  constexpr size_t smem5 = (128 * 10  * 10  + 1   * 128 * 16) * 4; //  59,392

  auto allow = [](const void* f, size_t s) {
    (void)hipFuncSetAttribute(f, hipFuncAttributeMaxDynamicSharedMemorySize, (int)s);
  };
  allow((const void*)adder2d_kernel<1,  4, 128, 128, 64, 64, 16,  16, 16>, smem1);
  allow((const void*)adder2d_kernel<16, 4, 64,  64,  32, 32, 32,  32, 16>, smem2);
  allow((const void*)adder2d_kernel<32, 3, 32,  32,  16, 16, 64,  64, 16>, smem3);
  allow((const void*)adder2d_kernel<64, 4, 16,  16,  8,  8,  128, 32, 8 >, smem4);
  allow((const void*)adder2d_kernel<128,4, 8,   8,   4,  4,  1,   1,  4 >, smem5);

  // ---- Layer 1: [N,1,128,128] -> [N,16,64,64] ----
  adder2d_kernel<1, 4, 128, 128, 64, 64, 16, 16, 16>
      <<<dim3(N, 1), blk, smem1, stream>>>(x, W1, bufA);
  bn_stats_kernel<<<dim3(16), blk, 0, stream>>>(bufA, g1, b1, stats, N, 16, 64 * 64);
  {
    long tot = (long)N * 16 * 64 * 64;
    bn_apply_lrelu_kernel<<<dim3((unsigned)((tot + 255) / 256)), blk, 0, stream>>>(
        bufA, stats, 16, 64 * 64, tot);
  }

  // ---- Layer 2: [N,16,64,64] -> [N,32,32,32] ----
  adder2d_kernel<16, 4, 64, 64, 32, 32, 32, 32, 16>
      <<<dim3(N, 1), blk, smem2, stream>>>(bufA, W2, bufB);
  bn_stats_kernel<<<dim3(32), blk, 0, stream>>>(bufB, g2, b2, stats, N, 32, 32 * 32);
  {
    long tot = (long)N * 32 * 32 * 32;
    bn_apply_lrelu_kernel<<<dim3((unsigned)((tot + 255) / 256)), blk, 0, stream>>>(
        bufB, stats, 32, 32 * 32, tot);
  }

  // ---- Layer 3: [N,32,32,32] -> [N,64,16,16] (k=3) ----
  adder2d_kernel<32, 3, 32, 32, 16, 16, 64, 64, 16>
      <<<dim3(N, 1), blk, smem3, stream>>>(bufB, W3, bufA);
  bn_stats_kernel<<<dim3(64), blk, 0, stream>>>(bufA, g3, b3, stats, N, 64, 16 * 16);
  {
    long tot = (long)N * 64 * 16 * 16;
    bn_apply_lrelu_kernel<<<dim3((unsigned)((tot + 255) / 256)), blk, 0, stream>>>(
        bufA, stats, 64, 16 * 16, tot);
  }

  // ---- Layer 4: [N,64,16,16] -> [N,128,8,8] (CO tiled by 32) ----
  adder2d_kernel<64, 4, 16, 16, 8, 8, 128, 32, 8>
      <<<dim3(N, 4), blk, smem4, stream>>>(bufA, W4, bufB);
  bn_stats_kernel<<<dim3(128), blk, 0, stream>>>(bufB, g4, b4, stats, N, 128, 8 * 8);
  {
    long tot = (long)N * 128 * 8 * 8;
    bn_apply_lrelu_kernel<<<dim3((unsigned)((tot + 255) / 256)), blk, 0, stream>>>(
        bufB, stats, 128, 8 * 8, tot);
  }

  // ---- Layer 5: [N,128,8,8] -> [N,1,4,4], BN + sigmoid -> d_out ----
  adder2d_kernel<128, 4, 8, 8, 4, 4, 1, 1, 4>
      <<<dim3(N, 1), blk, smem5, stream>>>(bufB, W5, bufA);
  bn_stats_kernel<<<dim3(1), blk, 0, stream>>>(bufA, g5, b5, stats, N, 1, 4 * 4);
  {
    long tot = (long)N * 1 * 4 * 4;
    bn_apply_sigmoid_kernel<<<dim3((unsigned)((tot + 255) / 256)), blk, 0, stream>>>(
        bufA, stats, out, 1, 4 * 4, tot);
  }

  (void)n_in; (void)out_size; (void)ws_size;
}